// CentroidLayer_24129126269560
// MI455X (gfx1250) — compile-verified
//
#include <hip/hip_runtime.h>
#include <hip/hip_bf16.h>

// Problem dims (from reference): B=256, NC=32, N_CH=8, P=32
#define B_    256
#define NC_   32
#define NCH_  8
#define P_    32
#define LDST  33   // padded LDS row stride (floats) to break bank conflicts

typedef __attribute__((ext_vector_type(2))) float v2f;
typedef __attribute__((ext_vector_type(8))) float v8f;

// ---------------------------------------------------------------------------
// Kernel 1: batched 32x32 cyclic-Jacobi eigendecomposition of C[cn] (SPD).
// One wave32 per matrix. Writes eigenvector matrices (columns sorted by
// ascending eigenvalue, matching jnp.linalg.eigh up to column sign — which
// the downstream quadratic form is invariant to) into d_ws.
// Layout out: Vout[cn][j][i]  (j = row, i = eigenvector index), cn = c*8+n.
// ---------------------------------------------------------------------------
__global__ __launch_bounds__(32)
void jacobi_eigh_kernel(const float* __restrict__ C, float* __restrict__ Vout) {
  __shared__ float A[P_ * LDST];
  __shared__ float V[P_ * LDST];
  const int cn   = blockIdx.x;
  const int lane = threadIdx.x;           // 0..31 (wave32)
  const float* Cm = C + (size_t)cn * P_ * P_;

  for (int r = 0; r < P_; ++r) {
    A[r * LDST + lane] = Cm[r * P_ + lane];
    V[r * LDST + lane] = (r == lane) ? 1.f : 0.f;
  }
  __syncthreads();

  for (int sweep = 0; sweep < 10; ++sweep) {
    for (int p = 0; p < P_ - 1; ++p) {
      for (int q = p + 1; q < P_; ++q) {
        const float apq = A[p * LDST + q];
        const float app = A[p * LDST + p];
        const float aqq = A[q * LDST + q];
        float c = 1.f, s = 0.f;
        if (fabsf(apq) > 1e-12f) {        // uniform across the wave
          const float tau = (aqq - app) / (2.f * apq);
          const float t   = copysignf(1.f, tau) /
                            (fabsf(tau) + sqrtf(1.f + tau * tau));
          c = 1.f / sqrtf(1.f + t * t);
          s = t * c;
        }
        const int k = lane;
        // A <- A*G  (rotate columns p,q)
        const float akp = A[k * LDST + p], akq = A[k * LDST + q];
        A[k * LDST + p] = c * akp - s * akq;
        A[k * LDST + q] = s * akp + c * akq;
        __syncthreads();                  // single-wave WG: lowers to no-op+wait
        // A <- G^T*A (rotate rows p,q)
        const float apk = A[p * LDST + k], aqk = A[q * LDST + k];
        A[p * LDST + k] = c * apk - s * aqk;
        A[q * LDST + k] = s * apk + c * aqk;
        // V <- V*G
        const float vkp = V[k * LDST + p], vkq = V[k * LDST + q];
        V[k * LDST + p] = c * vkp - s * vkq;
        V[k * LDST + q] = s * vkp + c * vkq;
        __syncthreads();
      }
    }
  }

  // Sort columns by ascending eigenvalue (rank via stable comparison).
  const int i  = lane;
  const float ei = A[i * LDST + i];
  int rank = 0;
  for (int j = 0; j < P_; ++j) {
    const float ej = A[j * LDST + j];
    rank += ((ej < ei) || (ej == ei && j < i)) ? 1 : 0;
  }
  float* Vo = Vout + (size_t)cn * P_ * P_;
  for (int r = 0; r < P_; ++r)
    Vo[r * P_ + rank] = V[r * LDST + i];
}

// ---------------------------------------------------------------------------
// Kernel 2: diag[b,c,n,i] = sum_j (X[b,n] @ V[c,n])[j,i] * V[c,n][j,i]
// Grid: x = cn (256), y = b-chunk (8). Block = 128 threads = 4 waves.
//
// X is SPD (symmetric), so the 16x4 f32 A-fragments can be fetched directly
// from global memory with coalesced addresses: A[m,k] = X[k][m], so lane l
// reads 64B-contiguous runs of row kb+v — no LDS transpose staging needed.
// V[c,n] is staged once into LDS; both the WMMA B-fragments and the
// diagonal-extraction weights are hoisted into registers before the b-loop,
// so the steady state per X matrix is:
//   32 coalesced global_load_b32 -> 32 v_wmma_f32_16x16x4_f32 -> 32 FMA + shfl
// ---------------------------------------------------------------------------
__global__ __launch_bounds__(128)
void centroid_diag_kernel(const float* __restrict__ X,
                          const float* __restrict__ Veig,
                          float* __restrict__ Out) {
  __shared__ float Vs[P_ * LDST];

  const int cn   = blockIdx.x;            // c*NCH + n
  const int c    = cn >> 3;
  const int n    = cn & 7;
  const int tid  = threadIdx.x;
  const int w    = tid >> 5;              // wave id 0..3
  const int lane = tid & 31;
  const int half = lane >> 4;             // 0: lanes 0-15, 1: lanes 16-31
  const int l16  = lane & 15;

  // Stage eigenvector matrix for this (c,n): 1024 floats, 128 threads.
  const float* Vg = Veig + (size_t)cn * P_ * P_;
  for (int it = 0; it < 8; ++it) {
    const int flat = it * 128 + tid;
    Vs[(flat >> 5) * LDST + (flat & 31)] = Vg[flat];
  }
  __syncthreads();                        // Vs read-only from here on

  // WMMA B-fragments (4x16 tiles of V) in registers, invariant over b.
  // B layout: lane l holds col l%16, K = 4*kk + 2*(l/16) + {0,1}.
  v2f bf0[8], bf1[8];
#pragma unroll
  for (int kk = 0; kk < 8; ++kk) {
    const int kb = 4 * kk + 2 * half;
    bf0[kk].x = Vs[kb * LDST + l16];       bf0[kk].y = Vs[(kb + 1) * LDST + l16];
    bf1[kk].x = Vs[kb * LDST + l16 + 16];  bf1[kk].y = Vs[(kb + 1) * LDST + l16 + 16];
  }

  // Diagonal-extraction weights V[j,i] in registers, invariant over b.
  // Accumulator layout: VGPR r, lane l -> row r + 8*(l/16), col l%16.
  const int rh = 8 * half;
  float dv0[16], dv1[16];
#pragma unroll
  for (int r = 0; r < 8; ++r) {
    dv0[r]     = Vs[(r + rh) * LDST + l16];            // mi=0 tile, ni=0
    dv0[8 + r] = Vs[(16 + r + rh) * LDST + l16];       // mi=1 tile, ni=0
    dv1[r]     = Vs[(r + rh) * LDST + l16 + 16];       // mi=0 tile, ni=1
    dv1[8 + r] = Vs[(16 + r + rh) * LDST + l16 + 16];  // mi=1 tile, ni=1
  }

  const int b0 = blockIdx.y * 32 + w * 8;
  for (int bi = 0; bi < 8; ++bi) {
    const int b = b0 + bi;
    const float* Xg = X + ((size_t)b * NCH_ + n) * P_ * P_;
    if (bi < 7)   // prefetch next X matrix -> global_prefetch_b8
      __builtin_prefetch(Xg + (size_t)NCH_ * P_ * P_, 0, 1);

    // W = X * V via f32 WMMA: 2x2 tiles of 16x16, K in steps of 4.
    v8f acc00 = {0.f,0.f,0.f,0.f,0.f,0.f,0.f,0.f};
    v8f acc01 = acc00, acc10 = acc00, acc11 = acc00;
#pragma unroll
    for (int kk = 0; kk < 8; ++kk) {
      const int kb = 4 * kk + 2 * half;
      v2f a0, a1;
      // A[m, k] = X[m][k] = X[k][m] (symmetry) -> coalesced row reads.
      a0.x = Xg[kb * P_ + l16];        a0.y = Xg[(kb + 1) * P_ + l16];
      a1.x = Xg[kb * P_ + l16 + 16];   a1.y = Xg[(kb + 1) * P_ + l16 + 16];
      acc00 = __builtin_amdgcn_wmma_f32_16x16x4_f32(false, a0, false, bf0[kk],
                                                    (short)0, acc00, false, false);
      acc01 = __builtin_amdgcn_wmma_f32_16x16x4_f32(false, a0, false, bf1[kk],
                                                    (short)0, acc01, false, false);
      acc10 = __builtin_amdgcn_wmma_f32_16x16x4_f32(false, a1, false, bf0[kk],
                                                    (short)0, acc10, false, false);
      acc11 = __builtin_amdgcn_wmma_f32_16x16x4_f32(false, a1, false, bf1[kk],
                                                    (short)0, acc11, false, false);
    }

    // diag[i] = sum_j W[j,i] * V[j,i]; fold row-halves with shfl_xor(16).
    float p0 = 0.f, p1 = 0.f;
#pragma unroll
    for (int r = 0; r < 8; ++r) {
      p0 += acc00[r] * dv0[r] + acc10[r] * dv0[8 + r];
      p1 += acc01[r] * dv1[r] + acc11[r] * dv1[8 + r];
    }
    p0 += __shfl_xor(p0, 16, 32);
    p1 += __shfl_xor(p1, 16, 32);
    if (half == 0) {
      float* o = Out + (((size_t)b * NC_ + c) * NCH_ + n) * P_;
      o[l16]      = p0;
      o[l16 + 16] = p1;
    }
  }
}

// ---------------------------------------------------------------------------
extern "C" void kernel_launch(void* const* d_in, const int* in_sizes, int n_in,
                              void* d_out, int out_size, void* d_ws, size_t ws_size,
                              hipStream_t stream) {
  (void)in_sizes; (void)n_in; (void)out_size; (void)ws_size;
  const float* X = (const float*)d_in[0];   // [B, NCH, P, P]
  const float* C = (const float*)d_in[1];   // [NC, NCH, P, P]
  // d_in[2] = idx : unused on the inference path.
  float* Vout = (float*)d_ws;               // [NC*NCH, P, P] = 1 MB scratch
  float* Out  = (float*)d_out;              // [B, NC, NCH, P]

  jacobi_eigh_kernel<<<dim3(NC_ * NCH_), dim3(32), 0, stream>>>(C, Vout);
  centroid_diag_kernel<<<dim3(NC_ * NCH_, B_ / 32), dim3(128), 0, stream>>>(X, Vout, Out);
}